// logits_loss_39676907880504
// MI455X (gfx1250) — compile-verified
//
#include <hip/hip_runtime.h>

typedef __attribute__((ext_vector_type(2))) float v2f;
typedef __attribute__((ext_vector_type(4))) float v4f;
typedef __attribute__((ext_vector_type(8))) float v8f;

static constexpr int   Ccls  = 32768;
static constexpr int   Dfeat = 1024;
static constexpr int   Bbat  = 16384;
static constexpr float KEEP  = 0.95f;

// ---------------- init ----------------
__global__ void k_zero(int* __restrict__ counts, float* __restrict__ out) {
  int i = blockIdx.x * blockDim.x + threadIdx.x;
  if (i < Ccls) counts[i] = 0;
  if (i == 0) out[0] = 0.0f;
}

// ---------------- histogram ----------------
__global__ void k_count(const int* __restrict__ cls, int* __restrict__ counts) {
  int i = blockIdx.x * blockDim.x + threadIdx.x;
  atomicAdd(&counts[cls[i]], 1);
}

// ---------------- per-sample EMA weight: w_i = (1-keep)*keep^{#later same-class} ----------------
__global__ void k_weights(const int* __restrict__ cls, const int* __restrict__ counts,
                          float* __restrict__ w) {
  __shared__ int tile[256];
  const int i   = blockIdx.x * 256 + threadIdx.x;
  const int c   = cls[i];
  const int cnt = counts[c];
  int r = 0;
  for (int base = blockIdx.x * 256; base < Bbat; base += 256) {
    tile[threadIdx.x] = cls[base + threadIdx.x];
    __syncthreads();
    if (cnt > 1) {
#pragma unroll 8
      for (int t2 = 0; t2 < 256; ++t2)
        r += ((base + t2 > i) && (tile[t2] == c)) ? 1 : 0;
    }
    __syncthreads();
  }
  w[i] = (1.0f - KEEP) * __powf(KEEP, (float)r);
}

// ---------------- exclusive scan of counts -> CSR offsets (+cursor copy) ----------------
__global__ void k_scan(const int* __restrict__ counts, int* __restrict__ offsets,
                       int* __restrict__ cursor) {
  __shared__ int sums[1024];
  const int t = threadIdx.x;
  constexpr int CHUNK = Ccls / 1024; // 32
  int local[CHUNK];
  int s = 0;
  const int base = t * CHUNK;
#pragma unroll
  for (int k = 0; k < CHUNK; ++k) { local[k] = s; s += counts[base + k]; }
  sums[t] = s;
  __syncthreads();
  for (int off = 1; off < 1024; off <<= 1) {
    int v = (t >= off) ? sums[t - off] : 0;
    __syncthreads();
    sums[t] += v;
    __syncthreads();
  }
  const int prefix = (t == 0) ? 0 : sums[t - 1];
#pragma unroll
  for (int k = 0; k < CHUNK; ++k) {
    int o = prefix + local[k];
    offsets[base + k] = o;
    cursor[base + k]  = o;
  }
  if (t == 1023) offsets[Ccls] = prefix + s; // == B
}

// ---------------- scatter sample indices into CSR buckets ----------------
__global__ void k_fill(const int* __restrict__ cls, int* __restrict__ cursor,
                       int* __restrict__ items) {
  int i   = blockIdx.x * blockDim.x + threadIdx.x;
  int pos = atomicAdd(&cursor[cls[i]], 1);
  items[pos] = i;
}

// ---------------- fused EMA + |.| row-sum + mean; one wave32 per class row ----------------
__global__ void __launch_bounds__(256) k_loss(
    const float* __restrict__ s, const float* __restrict__ t,
    const float* __restrict__ logits, const int* __restrict__ counts,
    const int* __restrict__ offsets, const int* __restrict__ items,
    const float* __restrict__ w, float* __restrict__ out) {
  const int lane = threadIdx.x & 31;
  const int wave = threadIdx.x >> 5;
  const int c    = blockIdx.x * 8 + wave;

  const float* srow = s + (size_t)c * Dfeat;
  const float* trow = t + (size_t)c * Dfeat;

  const float decay = __powf(KEEP, (float)counts[c]);
  const int beg = offsets[c];
  const int end = offsets[c + 1];

  if (end > beg) { // wave-uniform branch; warms L2 for the gather
    __builtin_prefetch(logits + (size_t)items[beg] * Dfeat + lane * 4, 0, 1);
  }

  // each lane owns 8 v4f chunks (32 values) of the 1024-wide row
  v4f acc[8];
#pragma unroll
  for (int k = 0; k < 8; ++k) {
    const v4f sv = __builtin_nontemporal_load((const v4f*)(srow + k * 128 + lane * 4));
    const v4f tv = __builtin_nontemporal_load((const v4f*)(trow + k * 128 + lane * 4));
    acc[k] = sv * decay - tv;
  }

  for (int it = beg; it < end; ++it) {   // wave-uniform loop bounds -> EXEC stays all-ones
    const int   i     = items[it];
    const float wi    = w[i];
    const float* lrow = logits + (size_t)i * Dfeat;
#pragma unroll
    for (int k = 0; k < 8; ++k) {
      const v4f lv = __builtin_nontemporal_load((const v4f*)(lrow + k * 128 + lane * 4));
      acc[k] += wi * lv;
    }
  }

  // abs-row-sum via v_wmma_f32_16x16x4_f32 against all-ones B:
  // D[m,n] = sum_k A[m,k]  =>  sum_m D[m,0] = sum of all 64 fed values (placement-invariant)
  v8f cacc = {};
  const v2f ones = {1.0f, 1.0f};
#pragma unroll
  for (int k = 0; k < 8; ++k) {
    v2f a0; a0[0] = __builtin_fabsf(acc[k][0]); a0[1] = __builtin_fabsf(acc[k][1]);
    v2f a1; a1[0] = __builtin_fabsf(acc[k][2]); a1[1] = __builtin_fabsf(acc[k][3]);
    cacc = __builtin_amdgcn_wmma_f32_16x16x4_f32(false, a0, false, ones,
                                                 (short)0, cacc, false, false);
    cacc = __builtin_amdgcn_wmma_f32_16x16x4_f32(false, a1, false, ones,
                                                 (short)0, cacc, false, false);
  }

  // column 0 lives on lane 0 (M=0..7 in cacc[0..7]) and lane 16 (M=8..15)
  float p = cacc[0] + cacc[1] + cacc[2] + cacc[3] + cacc[4] + cacc[5] + cacc[6] + cacc[7];
  const float rowsum = __shfl(p, 0, 32) + __shfl(p, 16, 32);
  if (lane == 0) atomicAdd(out, rowsum * (1.0f / (float)Ccls));
}

// ---------------- host ----------------
extern "C" void kernel_launch(void* const* d_in, const int* in_sizes, int n_in,
                              void* d_out, int out_size, void* d_ws, size_t ws_size,
                              hipStream_t stream) {
  (void)in_sizes; (void)n_in; (void)out_size; (void)ws_size;
  const float* s      = (const float*)d_in[0];
  const float* t      = (const float*)d_in[1];
  const float* logits = (const float*)d_in[2];
  const int*   cls    = (const int*)d_in[3];
  float*       out    = (float*)d_out;

  char* ws = (char*)d_ws;
  int*   counts  = (int*)(ws);                                   // 32768 ints
  int*   offsets = (int*)(ws + 128 * 1024);                      // 32769 ints
  int*   cursor  = (int*)(ws + 2 * 128 * 1024 + 1024);           // 32768 ints
  int*   items   = (int*)(ws + 3 * 128 * 1024 + 2048);           // 16384 ints
  float* w       = (float*)(ws + 3 * 128 * 1024 + 2048 + 64 * 1024); // 16384 f32

  k_zero   <<<(Ccls + 255) / 256, 256, 0, stream>>>(counts, out);
  k_count  <<<Bbat / 256, 256, 0, stream>>>(cls, counts);
  k_weights<<<Bbat / 256, 256, 0, stream>>>(cls, counts, w);
  k_scan   <<<1, 1024, 0, stream>>>(counts, offsets, cursor);
  k_fill   <<<Bbat / 256, 256, 0, stream>>>(cls, cursor, items);
  k_loss   <<<Ccls / 8, 256, 0, stream>>>(s, t, logits, counts, offsets, items, w, out);
}